// gat_1992864825774
// MI455X (gfx1250) — compile-verified
//
#include <hip/hip_runtime.h>

#define N_NODES  100000
#define IN_C     300
#define OUT_C    200
#define N_EDGES  1600000
#define E_TOTAL  (N_EDGES + N_NODES)   // self-loops appended
#define NEG_SLOPE 0.2f

typedef __attribute__((ext_vector_type(2))) float v2f;
typedef __attribute__((ext_vector_type(8))) float v8f;

// ---------------------------------------------------------------------------
// Tiled f32 GEMM using V_WMMA_F32_16X16X4_F32.
// C[M,N] = A[M,K] @ B[K,N], all row-major f32.
// Block: 256 threads (8 waves). Block tile 128(M) x 64(N), BK = 32.
// Wave w computes rows w*16..w*16+15, all 64 cols (4 WMMA n-subtiles).
// ISA layouts (CDNA5 ISA §7.12.2):
//   A 16x4 f32 : lane 0-15 -> M=lane, VGPRj -> K = 2*half + j
//   B 4x16 f32 : VGPRj, half h -> row K = 2*h + j, col = lane&15
//   C 16x16 f32: VGPRv -> M = v + 8*half, N = lane&15
// LDS layout tuned so every fragment is one 8B-aligned ds_load_b64:
//   As[row][AS_STRIDE] with even AS_STRIDE=34 -> (K,K+1) pair aligned
//   Bs interleaved [k/2][col][k&1]           -> (rowK,rowK+1) pair contiguous
// ---------------------------------------------------------------------------
#define BM 128
#define BN 64
#define BK 32
#define AS_STRIDE 34   // EVEN (b64-aligned frags) + odd/2 factor dodges bank conflicts

__global__ __launch_bounds__(256)
void gemm_wmma_f32(const float* __restrict__ A, const float* __restrict__ B,
                   float* __restrict__ C, int M, int K, int N) {
  __shared__ float As[BM][AS_STRIDE];
  __shared__ float Bs[BK / 2][BN][2];   // [k-pair][col][parity]

  const int tid  = threadIdx.x;
  const int wave = tid >> 5;
  const int lane = tid & 31;
  const int l16  = lane & 15;
  const int half = lane >> 4;
  const int m0   = blockIdx.x * BM;
  const int n0   = blockIdx.y * BN;
  const int wrow = wave * 16;

  v8f acc0 = {}, acc1 = {}, acc2 = {}, acc3 = {};

  for (int k0 = 0; k0 < K; k0 += BK) {
    // ---- stage A tile (128 x 32): 256 threads, float4 each, 4 row passes
    {
      const int kk = (tid & 7) * 4;
      const int gk = k0 + kk;
      for (int rp = 0; rp < BM; rp += 32) {
        const int r    = rp + (tid >> 3);
        const int grow = m0 + r;
        float4 v = make_float4(0.f, 0.f, 0.f, 0.f);
        if (grow < M) {
          const float* ap = A + (size_t)grow * K;
          if (gk + 3 < K) {
            v = *(const float4*)(ap + gk);        // 16B-aligned: K%4==0, k0%32==0
          } else {
            if (gk + 0 < K) v.x = ap[gk + 0];
            if (gk + 1 < K) v.y = ap[gk + 1];
            if (gk + 2 < K) v.z = ap[gk + 2];
            if (gk + 3 < K) v.w = ap[gk + 3];
          }
        }
        *(float2*)&As[r][kk + 0] = make_float2(v.x, v.y);   // 8B aligned (even stride)
        *(float2*)&As[r][kk + 2] = make_float2(v.z, v.w);
      }
    }
    // ---- stage B tile (32 x 64) interleaved: thread handles one (k-pair, col)
    for (int i = tid; i < (BK / 2) * BN; i += 256) {
      const int kp = i >> 6;          // 0..15
      const int c  = i & 63;          // 0..63
      const int gk = k0 + 2 * kp;
      const int gc = n0 + c;
      float2 bv = make_float2(0.f, 0.f);
      if (gc < N) {
        if (gk + 0 < K) bv.x = B[(size_t)(gk + 0) * N + gc];
        if (gk + 1 < K) bv.y = B[(size_t)(gk + 1) * N + gc];
      }
      *(float2*)&Bs[kp][c][0] = bv;   // stride-2-word across lanes: conflict-free
    }
    __syncthreads();

    // ---- compute: 8 K-steps of 4, four 16-wide N subtiles each
    #pragma unroll
    for (int kk = 0; kk < BK; kk += 4) {
      const int kp = (kk >> 1) + half;                       // B k-pair for this lane
      const v2f a  = *(const v2f*)&As[wrow + l16][kk + 2 * half];
      const v2f b0 = *(const v2f*)&Bs[kp][ 0 + l16][0];
      const v2f b1 = *(const v2f*)&Bs[kp][16 + l16][0];
      const v2f b2 = *(const v2f*)&Bs[kp][32 + l16][0];
      const v2f b3 = *(const v2f*)&Bs[kp][48 + l16][0];
      acc0 = __builtin_amdgcn_wmma_f32_16x16x4_f32(false, a, false, b0, (short)0, acc0, false, false);
      acc1 = __builtin_amdgcn_wmma_f32_16x16x4_f32(false, a, false, b1, (short)0, acc1, false, false);
      acc2 = __builtin_amdgcn_wmma_f32_16x16x4_f32(false, a, false, b2, (short)0, acc2, false, false);
      acc3 = __builtin_amdgcn_wmma_f32_16x16x4_f32(false, a, false, b3, (short)0, acc3, false, false);
    }
    __syncthreads();
  }

  // ---- store C: VGPR v -> row = v + 8*half, col = subtile*16 + l16
  #pragma unroll
  for (int t = 0; t < 4; t++) {
    const int col = n0 + t * 16 + l16;
    if (col >= N) continue;
    const v8f acc = (t == 0) ? acc0 : (t == 1) ? acc1 : (t == 2) ? acc2 : acc3;
    #pragma unroll
    for (int v = 0; v < 8; v++) {
      const int row = m0 + wrow + v + 8 * half;
      if (row < M) C[(size_t)row * N + col] = acc[v];
    }
  }
}

// ---------------------------------------------------------------------------
// Per-node attention logits: asrc[n] = h[n]·a_src, adst[n] = h[n]·a_dst
// One wave32 per node, shuffle reduction.
// ---------------------------------------------------------------------------
__global__ __launch_bounds__(256)
void alpha_kernel(const float* __restrict__ h, const float* __restrict__ a_src,
                  const float* __restrict__ a_dst, float* __restrict__ asrc,
                  float* __restrict__ adst) {
  const int node = blockIdx.x * 8 + (threadIdx.x >> 5);
  const int lane = threadIdx.x & 31;
  if (node >= N_NODES) return;
  const float* hr = h + (size_t)node * OUT_C;
  float s0 = 0.f, s1 = 0.f;
  for (int c = lane; c < OUT_C; c += 32) {
    const float v = hr[c];
    s0 += v * a_src[c];
    s1 += v * a_dst[c];
  }
  #pragma unroll
  for (int off = 16; off > 0; off >>= 1) {
    s0 += __shfl_xor(s0, off, 32);
    s1 += __shfl_xor(s1, off, 32);
  }
  if (lane == 0) { asrc[node] = s0; adst[node] = s1; }
}

// Sortable-uint encoding of f32 so segment-max can use u32 atomicMax.
__device__ __forceinline__ unsigned f2sort(float f) {
  unsigned b = __float_as_uint(f);
  return (b & 0x80000000u) ? ~b : (b | 0x80000000u);
}
__device__ __forceinline__ float sort2f(unsigned u) {
  unsigned b = (u & 0x80000000u) ? (u & 0x7FFFFFFFu) : ~u;
  return __uint_as_float(b);
}

__device__ __forceinline__ void edge_sd(const long long* __restrict__ ei,
                                        int idx, int& s, int& d) {
  if (idx < N_EDGES) { s = (int)ei[idx]; d = (int)ei[N_EDGES + idx]; }
  else               { s = d = idx - N_EDGES; }          // self-loop
}

// e = leakyrelu(asrc[s]+adst[d]); stash e; segment-max via u32 atomicMax
__global__ __launch_bounds__(256)
void edge_max_kernel(const long long* __restrict__ ei,
                     const float* __restrict__ asrc, const float* __restrict__ adst,
                     float* __restrict__ ebuf, unsigned* __restrict__ mbuf) {
  const int idx = blockIdx.x * blockDim.x + threadIdx.x;
  if (idx >= E_TOTAL) return;
  int s, d; edge_sd(ei, idx, s, d);
  float e = asrc[s] + adst[d];
  e = (e > 0.f) ? e : NEG_SLOPE * e;
  ebuf[idx] = e;
  atomicMax(mbuf + d, f2sort(e));   // encoded reals are all > 0 => init 0 == -inf
}

// ex = exp(e - m[d]); overwrite ebuf with ex; denom[d] += ex
__global__ __launch_bounds__(256)
void edge_exp_kernel(const long long* __restrict__ ei, float* __restrict__ ebuf,
                     const unsigned* __restrict__ mbuf, float* __restrict__ denom) {
  const int idx = blockIdx.x * blockDim.x + threadIdx.x;
  if (idx >= E_TOTAL) return;
  int s, d; edge_sd(ei, idx, s, d);
  const float ex = __expf(ebuf[idx] - sort2f(mbuf[d]));
  ebuf[idx] = ex;
  atomicAdd(denom + d, ex);
}

// acc[d] += h[s] * (ex/denom[d]) : one wave32 per edge, f32 atomics into L2
__global__ __launch_bounds__(256)
void edge_scatter_kernel(const long long* __restrict__ ei,
                         const float* __restrict__ ebuf, const float* __restrict__ denom,
                         const float* __restrict__ h, float* __restrict__ acc) {
  const int widx = blockIdx.x * 8 + (threadIdx.x >> 5);
  const int lane = threadIdx.x & 31;
  if (widx >= E_TOTAL) return;
  int s, d; edge_sd(ei, widx, s, d);
  const float alpha = ebuf[widx] / denom[d];
  const float* hs = h + (size_t)s * OUT_C;
  float* ad = acc + (size_t)d * OUT_C;
  for (int c = lane; c < OUT_C; c += 32)
    atomicAdd(ad + c, hs[c] * alpha);
}

// y = relu(acc + bias), in place
__global__ __launch_bounds__(256)
void finalize_kernel(float* __restrict__ acc, const float* __restrict__ bias) {
  const int i = blockIdx.x * blockDim.x + threadIdx.x;
  if (i >= N_NODES * OUT_C) return;
  const float v = acc[i] + bias[i % OUT_C];
  acc[i] = (v > 0.f) ? v : 0.f;
}

// ---------------------------------------------------------------------------
extern "C" void kernel_launch(void* const* d_in, const int* in_sizes, int n_in,
                              void* d_out, int out_size, void* d_ws, size_t ws_size,
                              hipStream_t stream) {
  const float*     x      = (const float*)d_in[0];
  const float*     W0     = (const float*)d_in[1];
  const float*     a_src0 = (const float*)d_in[2];
  const float*     a_dst0 = (const float*)d_in[3];
  const float*     b0     = (const float*)d_in[4];
  const float*     W1     = (const float*)d_in[5];
  const float*     a_src1 = (const float*)d_in[6];
  const float*     a_dst1 = (const float*)d_in[7];
  const float*     b1     = (const float*)d_in[8];
  const long long* ei     = (const long long*)d_in[9];   // int64 edge_index[2][E]
  float*           out    = (float*)d_out;

  char* ws = (char*)d_ws;
  float*    h     = (float*)(ws);                        //  80 MB: projected feats
  float*    acc   = (float*)(ws + 80000000);             //  80 MB: layer-1 aggregate
  float*    ebuf  = (float*)(ws + 160000000);            // 6.8 MB: per-edge e / ex
  float*    asrc  = (float*)(ws + 166800000);
  float*    adst  = (float*)(ws + 167200000);
  unsigned* mbuf  = (unsigned*)(ws + 167600000);
  float*    denom = (float*)(ws + 168000000);            // end: 168.4 MB

  const dim3 blk(256);
  const dim3 gemm_grid((N_NODES + BM - 1) / BM, (OUT_C + BN - 1) / BN);
  const int  node_wave_blocks = (N_NODES + 7) / 8;
  const int  edge_blocks      = (E_TOTAL + 255) / 256;
  const int  edge_wave_blocks = (E_TOTAL + 7) / 8;
  const int  elem_blocks      = (N_NODES * OUT_C + 255) / 256;

  // ---------------- layer 1 ----------------
  hipMemsetAsync(mbuf, 0, (size_t)N_NODES * 4, stream);   // 0 == encoded -inf
  hipMemsetAsync(denom, 0, (size_t)N_NODES * 4, stream);
  hipMemsetAsync(acc, 0, (size_t)N_NODES * OUT_C * 4, stream);

  gemm_wmma_f32<<<gemm_grid, blk, 0, stream>>>(x, W0, h, N_NODES, IN_C, OUT_C);
  alpha_kernel<<<node_wave_blocks, blk, 0, stream>>>(h, a_src0, a_dst0, asrc, adst);
  edge_max_kernel<<<edge_blocks, blk, 0, stream>>>(ei, asrc, adst, ebuf, mbuf);
  edge_exp_kernel<<<edge_blocks, blk, 0, stream>>>(ei, ebuf, mbuf, denom);
  edge_scatter_kernel<<<edge_wave_blocks, blk, 0, stream>>>(ei, ebuf, denom, h, acc);
  finalize_kernel<<<elem_blocks, blk, 0, stream>>>(acc, b0);

  // ---------------- layer 2 ----------------
  hipMemsetAsync(mbuf, 0, (size_t)N_NODES * 4, stream);
  hipMemsetAsync(denom, 0, (size_t)N_NODES * 4, stream);
  hipMemsetAsync(out, 0, (size_t)N_NODES * OUT_C * 4, stream);

  gemm_wmma_f32<<<gemm_grid, blk, 0, stream>>>(acc, W1, h, N_NODES, OUT_C, OUT_C);
  alpha_kernel<<<node_wave_blocks, blk, 0, stream>>>(h, a_src1, a_dst1, asrc, adst);
  edge_max_kernel<<<edge_blocks, blk, 0, stream>>>(ei, asrc, adst, ebuf, mbuf);
  edge_exp_kernel<<<edge_blocks, blk, 0, stream>>>(ei, ebuf, mbuf, denom);
  edge_scatter_kernel<<<edge_wave_blocks, blk, 0, stream>>>(ei, ebuf, denom, h, out);
  finalize_kernel<<<elem_blocks, blk, 0, stream>>>(out, b1);
}